// Large_batch_queue_8031588843986
// MI455X (gfx1250) — compile-verified
//
#include <hip/hip_runtime.h>
#include <cstdint>
#include <cstddef>

// ---------------------------------------------------------------------------
// Problem constants (match reference)
// ---------------------------------------------------------------------------
#define C_NUM 5532            // NUM_CLASSES
#define F_DIM 256             // feature dim
#define Q_SIZE (C_NUM * 2)    // queue rows = 11064

// TDM tiling: each wave double-buffers tiles of 4 rows x 256 f32 (4 KB)
#define TILE_ROWS   4
#define TILE_FLOATS (TILE_ROWS * F_DIM)   // 1024 floats = 4 KB
#define WAVES_PER_BLOCK 8                  // blockDim = 256

typedef unsigned int v4u __attribute__((ext_vector_type(4)));
typedef int          v8i __attribute__((ext_vector_type(8)));
typedef int          v4i __attribute__((ext_vector_type(4)));

// ---------------------------------------------------------------------------
// Kernel 1: init workspace (sums+counts -> 0.0f, inverse map -> -1)
// ---------------------------------------------------------------------------
__global__ void zero_ws(float* fzero, int nf, int* inv, int ni) {
    int i = blockIdx.x * blockDim.x + threadIdx.x;
    int stride = gridDim.x * blockDim.x;
    for (int k = i; k < nf; k += stride) fzero[k] = 0.0f;
    for (int k = i; k < ni; k += stride) inv[k] = -1;
}

// ---------------------------------------------------------------------------
// Kernel 2: per-class counts (float, to match reference segment_sum of ones)
// ---------------------------------------------------------------------------
__global__ void count_rows(const int* __restrict__ labels, float* __restrict__ counts, int n) {
    int i = blockIdx.x * blockDim.x + threadIdx.x;
    if (i < n) {
        unsigned l = (unsigned)labels[i];
        if (l < (unsigned)C_NUM) atomicAdd(&counts[l], 1.0f);
    }
}

// ---------------------------------------------------------------------------
// Kernel 3: segment sums. TDM (tensor_load_to_lds) stages 4-row feature
// tiles into wave-private LDS double buffers; lanes drain tiles from LDS as
// float4 and reduce into L2-resident accumulators with global_atomic_add_f32.
// D# built per CDNA5 ISA ch.8 (2-D tensor, data_size=4B).
// ---------------------------------------------------------------------------
__device__ __forceinline__ void tdm_issue_tile(const float* feats, int tile,
                                               unsigned lds_byte_addr) {
    unsigned long long ga = (unsigned long long)(uintptr_t)feats
                          + (unsigned long long)tile * (TILE_ROWS * F_DIM * 4ull);
    // ---- D# group 0 (128b): count=1 | lds_addr | global_addr | type=2 ----
    v4u g0;
    g0[0] = 1u;                                              // count = 1, user mode
    g0[1] = lds_byte_addr;                                   // lds_addr [63:32]
    g0[2] = (unsigned)(ga & 0xFFFFFFFFull);                  // global_addr[31:0]
    g0[3] = (unsigned)((ga >> 32) & 0x01FFFFFFull)           // global_addr[56:32]
          | (2u << 30);                                      // type = 2 ("image")
    // ---- D# group 1 (256b): 2-D tensor, 4-byte elements ----
    v8i g1;
    g1[0] = (2 << 16);                    // workgroup_mask=0, data_size=2 (4B)
    g1[1] = (F_DIM << 16);                // tensor_dim0 lo16 @ bits 63:48 (=256)
    g1[2] = (TILE_ROWS << 16);            // dim0 hi16=0; tensor_dim1 lo16 @ 95:80 (=4)
    g1[3] = (F_DIM << 16);                // dim1 hi16=0; tile_dim0 @ 127:112 (=256)
    g1[4] = TILE_ROWS;                    // tile_dim1 @ 143:128 (=4); tile_dim2=0
    g1[5] = F_DIM;                        // tensor_dim0_stride lo32 (=256 elements)
    g1[6] = 0;                            // stride hi16; tensor_dim1_stride lo16
    g1[7] = 0;                            // tensor_dim1_stride hi32
    v4i gz4 = {0, 0, 0, 0};               // groups 2/3 unused (2-D tensor)
    v8i gz8 = {0, 0, 0, 0, 0, 0, 0, 0};   // trailing operand (clang-23 6-arg form)
    __builtin_amdgcn_tensor_load_to_lds(g0, g1, gz4, gz4, gz8, 0);
}

__global__ void __launch_bounds__(WAVES_PER_BLOCK * 32)
segsum_tdm(const float* __restrict__ feats,
           const int* __restrict__ labels,
           float* __restrict__ sums, int n_rows) {
    __shared__ float lds_f[WAVES_PER_BLOCK * 2 * TILE_FLOATS];   // 64 KB

    const int lane = threadIdx.x & 31;
    const int wid  = threadIdx.x >> 5;
    const int wavesPerBlock = blockDim.x >> 5;
    const int gw = blockIdx.x * wavesPerBlock + wid;     // global wave id (uniform)
    const int nw = gridDim.x * wavesPerBlock;
    const int numTiles = n_rows / TILE_ROWS;

    // LDS byte address of this wave's buffers (flat addr low 32 bits == LDS byte addr)
    const unsigned ldsBase =
        (unsigned)(uintptr_t)(&lds_f[wid * 2 * TILE_FLOATS]);

    int tile = gw;
    if (tile < numTiles) tdm_issue_tile(feats, tile, ldsBase);   // prologue (uniform branch)
    int buf = 0;

    for (; tile < numTiles; tile += nw) {
        const int next = tile + nw;
        if (next < numTiles) {                                   // uniform per wave
            tdm_issue_tile(feats, next, ldsBase + (unsigned)((buf ^ 1) * TILE_FLOATS * 4));
            __builtin_amdgcn_s_wait_tensorcnt(1);                // current tile landed
        } else {
            __builtin_amdgcn_s_wait_tensorcnt(0);
        }

        const float4* tsrc =
            (const float4*)&lds_f[wid * 2 * TILE_FLOATS + buf * TILE_FLOATS];
        const int rowBase = tile * TILE_ROWS;

        #pragma unroll
        for (int j = 0; j < (TILE_FLOATS / 4) / 32; ++j) {       // 8 float4 per lane
            const int qq  = lane + j * 32;                       // quad index in tile
            const int r   = qq >> 6;                             // row within tile
            const int col = (qq & 63) * 4;
            float4 v = tsrc[qq];                                 // ds_load_b128
            unsigned l = (unsigned)labels[rowBase + r];          // L2-hot (512 KB array)
            if (l < (unsigned)C_NUM) {
                float* dst = sums + (size_t)l * F_DIM + col;
                atomicAdd(dst + 0, v.x);                         // global_atomic_add_f32
                atomicAdd(dst + 1, v.y);                         // (no-return form)
                atomicAdd(dst + 2, v.z);
                atomicAdd(dst + 3, v.w);
            }
        }
        buf ^= 1;
    }
}

// ---------------------------------------------------------------------------
// Kernel 4: rank of present classes (block-wide scan, wave32 ballot+popc)
// and scatter of inverse map inv[(tail+rank) % Q] = class.
// ---------------------------------------------------------------------------
__global__ void rank_scatter(const float* __restrict__ counts,
                             const int* __restrict__ tailp,
                             int* __restrict__ inv) {
    __shared__ unsigned wsum[32];
    __shared__ unsigned base;
    const int t = threadIdx.x, lane = t & 31, wid = t >> 5;
    if (t == 0) base = 0;
    __syncthreads();
    const long long tail = (long long)(*tailp);

    for (int start = 0; start < C_NUM; start += 1024) {
        const int c = start + t;
        const bool present = (c < C_NUM) && (counts[c] > 0.0f);
        const unsigned b32 = (unsigned)__ballot(present);        // wave32: low 32 bits
        const unsigned before = __popc(b32 & ((1u << lane) - 1u));
        if (lane == 0) wsum[wid] = __popc(b32);
        __syncthreads();
        if (wid == 0) {                                          // scan 32 wave totals
            unsigned v = wsum[lane];
            for (int off = 1; off < 32; off <<= 1) {
                unsigned u = __shfl_up(v, off, 32);
                if (lane >= off) v += u;
            }
            wsum[lane] = v;                                      // inclusive scan
        }
        __syncthreads();
        const unsigned rank = base + (wid ? wsum[wid - 1] : 0u) + before;
        if (present) {
            long long p = (tail + (long long)rank) % (long long)Q_SIZE;
            if (p < 0) p += Q_SIZE;
            inv[(int)p] = c;
        }
        __syncthreads();
        if (t == 0) base += wsum[31];
        __syncthreads();
    }
}

// ---------------------------------------------------------------------------
// Kernel 5: emit outputs. One block (64 threads, float4 lanes) per queue row.
// Writes every element of d_out (queue then labels).
// ---------------------------------------------------------------------------
__global__ void emit(const float* __restrict__ sums,
                     const float* __restrict__ counts,
                     const int* __restrict__ inv,
                     const float* __restrict__ queue,
                     const float* __restrict__ qlabel,
                     float* __restrict__ out) {
    const int q = blockIdx.x;
    const int t = threadIdx.x;                // 0..63, one float4 each
    const int c = inv[q];
    float4 v;
    if (c >= 0) {
        float cnt = counts[c];
        cnt = cnt > 1.0f ? cnt : 1.0f;
        float4 s = ((const float4*)(sums + (size_t)c * F_DIM))[t];
        v.x = s.x / cnt; v.y = s.y / cnt; v.z = s.z / cnt; v.w = s.w / cnt;
        if (t == 0) out[(size_t)Q_SIZE * F_DIM + q] = (float)c;
    } else {
        v = ((const float4*)(queue + (size_t)q * F_DIM))[t];
        if (t == 0) out[(size_t)Q_SIZE * F_DIM + q] = qlabel[q];
    }
    ((float4*)(out + (size_t)q * F_DIM))[t] = v;
}

// ---------------------------------------------------------------------------
// Host launcher
// ---------------------------------------------------------------------------
extern "C" void kernel_launch(void* const* d_in, const int* in_sizes, int n_in,
                              void* d_out, int out_size, void* d_ws, size_t ws_size,
                              hipStream_t stream) {
    const float* feats  = (const float*)d_in[0];   // [N, 256]
    const int*   labels = (const int*)d_in[1];     // [N]
    const float* queue  = (const float*)d_in[2];   // [Q, 256]
    const float* qlabel = (const float*)d_in[3];   // [Q]
    const int*   tailp  = (const int*)d_in[4];     // scalar
    float*       out    = (float*)d_out;           // Q*256 + Q floats

    const int N = in_sizes[1];

    // workspace layout: sums [C*F] f32 | counts [C] f32 | inv [Q] i32  (~5.8 MB)
    float* sums   = (float*)d_ws;
    float* counts = sums + (size_t)C_NUM * F_DIM;
    int*   inv    = (int*)(counts + C_NUM);

    const int nf = C_NUM * F_DIM + C_NUM;          // floats to zero
    zero_ws<<<2048, 256, 0, stream>>>(sums, nf, inv, Q_SIZE);

    count_rows<<<(N + 255) / 256, 256, 0, stream>>>(labels, counts, N);

    segsum_tdm<<<1024, WAVES_PER_BLOCK * 32, 0, stream>>>(feats, labels, sums, N);

    rank_scatter<<<1, 1024, 0, stream>>>(counts, tailp, inv);

    emit<<<Q_SIZE, F_DIM / 4, 0, stream>>>(sums, counts, inv, queue, qlabel, out);
}